// Decoder_48833778155531
// MI455X (gfx1250) — compile-verified
//
#include <hip/hip_runtime.h>
#include <math.h>

// ---------------------------------------------------------------------------
// Decoder.freerun on MI455X (gfx1250): persistent per-16-row-workgroup GRU
// recurrence using V_WMMA_F32_16X16X4_F32. Weights pre-packed into per-lane
// WMMA B-fragment order (float4 / lane / 2 WMMAs). fp32 throughout.
// ---------------------------------------------------------------------------

typedef float v2f __attribute__((ext_vector_type(2)));
typedef float v8f __attribute__((ext_vector_type(8)));

#define T_STEPS 64
#define VOCAB   256
#define LATD    256
#define HID     512
#define G3      1536   // 3*HID

__device__ __forceinline__ v8f wmma4(v2f a, v2f b, v8f c) {
  // D = A(16x4 f32) * B(4x16 f32) + C(16x16 f32)
  return __builtin_amdgcn_wmma_f32_16x16x4_f32(false, a, false, b,
                                               (short)0, c, false, false);
}

// Accumulate one 16x16 tile over K = ncp*8.
//   Bp   : packed tile base, layout [ncp][32 lanes][4 floats]
//   Arow : LDS row for this lane's M (stride = ncp*8 floats)
//   off  : (lane>=16) ? 2 : 0
__device__ __forceinline__ v8f mma_pass(v8f acc, const float* __restrict__ Bp,
                                        const float* Arow, int ncp, int off,
                                        int lane) {
  const float4* bp4 = reinterpret_cast<const float4*>(Bp) + lane;
#pragma unroll 8
  for (int c = 0; c < ncp; ++c) {
    float4 w = bp4[c * 32];                       // global_load_b128 (coalesced)
    float2 a0 = *reinterpret_cast<const float2*>(Arow + c * 8 + off);      // ds_load_b64
    float2 a1 = *reinterpret_cast<const float2*>(Arow + c * 8 + 4 + off);  // ds_load_b64
    v2f A0 = {a0.x, a0.y}, A1 = {a1.x, a1.y};
    v2f B0 = {w.x, w.y},   B1 = {w.z, w.w};
    acc = wmma4(A0, B0, acc);
    acc = wmma4(A1, B1, acc);
  }
  return acc;
}

__device__ __forceinline__ float sigf(float x) {
  return 1.0f / (1.0f + __expf(-x));
}

// ---------------------------------------------------------------------------
// Prep: pack W[N][K] (row-major, rowStride, colOff) into WMMA B-fragment order:
//   dst[((tile*ncp + c)*32 + lane)*4 + j] = W[tile*16 + (lane&15)][k(c,lane,j)]
// ---------------------------------------------------------------------------
__global__ void pack_frag(const float* __restrict__ W, float* __restrict__ dst,
                          int ntiles, int ncp, int rowStride, int colOff) {
  int total = ntiles * ncp * 128;
  int i = blockIdx.x * blockDim.x + threadIdx.x;
  if (i >= total) return;
  int j    = i & 3;
  int lane = (i >> 2) & 31;
  int c    = (i >> 7) % ncp;
  int t    = i / (ncp * 128);
  int n    = t * 16 + (lane & 15);
  int off  = (lane >> 4) << 1;
  int k    = c * 8 + ((j & 2) ? 4 : 0) + off + (j & 1);
  dst[i] = W[(size_t)n * rowStride + colOff + k];
}

// WohT[v][n] = w_ih0[n][256 + v]  (one-hot column table, row-contiguous in n)
__global__ void transpose_oh(const float* __restrict__ Wih0,
                             float* __restrict__ dst) {
  int i = blockIdx.x * blockDim.x + threadIdx.x;
  if (i >= VOCAB * G3) return;
  int v = i / G3, n = i % G3;
  dst[i] = Wih0[(size_t)n * 512 + LATD + v];
}

// ---------------------------------------------------------------------------
// Main persistent kernel: 64 blocks x 256 threads (8 waves), 16 rows each.
// ---------------------------------------------------------------------------
__global__ __launch_bounds__(256) void decoder_main(
    const float* __restrict__ latent,
    const float* __restrict__ bih0, const float* __restrict__ bhh0,
    const float* __restrict__ bih1, const float* __restrict__ bhh1,
    const float* __restrict__ bfc,
    const float* __restrict__ Pih0,  // packed w_ih0[:, :256], 96 tiles, ncp=32
    const float* __restrict__ WohT,  // 256 x 1536
    const float* __restrict__ Phh0,  // 96 tiles, ncp=64
    const float* __restrict__ Pih1,  // 96 tiles, ncp=64
    const float* __restrict__ Phh1,  // 96 tiles, ncp=64
    const float* __restrict__ Pfc,   // 16 tiles, ncp=64
    float* __restrict__ gi0ws,       // [64 wg][16][1536]
    float* __restrict__ out)         // [B][T][V]
{
  extern __shared__ float smem[];
  float* h0     = smem;                 // 16*512
  float* h1     = smem + 16 * HID;      // 16*512
  float* logits = smem + 2 * 16 * HID;  // 16*256
  int*   idxm   = (int*)(logits + 16 * VOCAB);  // 16

  const int tid  = threadIdx.x;
  const int lane = tid & 31;
  const int wave = tid >> 5;
  const int nl   = lane & 15;           // N within tile / M for A rows
  const int off  = (lane >> 4) << 1;    // K sub-offset for hi half-wave
  const int hi8  = (lane >> 4) << 3;    // M offset for C/D layout
  const int rowbase = blockIdx.x * 16;
  float* gi0 = gi0ws + (size_t)blockIdx.x * 16 * G3;

  // ---- init: stage latent rows in h1 region (stride 256) ----
  for (int i = tid; i < 16 * LATD; i += 256) {
    int m = i >> 8, k = i & 255;
    h1[m * LATD + k] = latent[(size_t)(rowbase + m) * LATD + k];
  }
  __syncthreads();
  // gi0 = latent @ w_ih0[:, :256].T + b_ih0   (96 tiles, 12 per wave)
  for (int j = 0; j < 12; ++j) {
    int t = wave * 12 + j;
    v8f acc = {};
    acc = mma_pass(acc, Pih0 + (size_t)t * 32 * 128, h1 + nl * LATD, 32, off, lane);
    float bias = bih0[t * 16 + nl];
#pragma unroll
    for (int e = 0; e < 8; ++e)
      gi0[(size_t)(e + hi8) * G3 + t * 16 + nl] = acc[e] + bias;
  }
  __syncthreads();
  for (int i = tid; i < 16 * HID; i += 256) { h0[i] = 0.f; h1[i] = 0.f; }
  if (tid < 16) idxm[tid] = -1;  // initial one-hot slot is all zeros
  __syncthreads();

  float hnew[4][8];

  for (int step = 0; step < T_STEPS; ++step) {
    // ================= layer 0 =================
#pragma unroll
    for (int j = 0; j < 4; ++j) {
      int t = wave + j * 8;  // output tile in [0,32)
      float ir[8], iz[8], in_[8];
#pragma unroll
      for (int e = 0; e < 8; ++e) {
        int m = e + hi8;
        const float* g = gi0 + (size_t)m * G3 + t * 16 + nl;
        float a0 = g[0], a1 = g[HID], a2 = g[2 * HID];
        int iv = idxm[m];
        if (iv >= 0) {
          const float* o = WohT + (size_t)iv * G3 + t * 16 + nl;
          a0 += o[0]; a1 += o[HID]; a2 += o[2 * HID];
        }
        ir[e] = a0; iz[e] = a1; in_[e] = a2;
      }
      v8f ar = {}, az = {}, an = {};
      ar = mma_pass(ar, Phh0 + (size_t)(t)      * 64 * 128, h0 + nl * HID, 64, off, lane);
      az = mma_pass(az, Phh0 + (size_t)(32 + t) * 64 * 128, h0 + nl * HID, 64, off, lane);
      an = mma_pass(an, Phh0 + (size_t)(64 + t) * 64 * 128, h0 + nl * HID, 64, off, lane);
      float brh = bhh0[t * 16 + nl];
      float bzh = bhh0[HID + t * 16 + nl];
      float bnh = bhh0[2 * HID + t * 16 + nl];
#pragma unroll
      for (int e = 0; e < 8; ++e) {
        float r  = sigf(ir[e] + ar[e] + brh);
        float z  = sigf(iz[e] + az[e] + bzh);
        float nn = tanhf(in_[e] + r * (an[e] + bnh));
        float ho = h0[(e + hi8) * HID + t * 16 + nl];
        hnew[j][e] = (1.f - z) * nn + z * ho;
      }
    }
    __syncthreads();
#pragma unroll
    for (int j = 0; j < 4; ++j) {
      int t = wave + j * 8;
#pragma unroll
      for (int e = 0; e < 8; ++e)
        h0[(e + hi8) * HID + t * 16 + nl] = hnew[j][e];
    }
    __syncthreads();

    // ================= layer 1 (x = new h0) =================
#pragma unroll
    for (int j = 0; j < 4; ++j) {
      int t = wave + j * 8;
      v8f ar = {}, az = {}, ain = {}, ahn = {};
      ar  = mma_pass(ar,  Pih1 + (size_t)(t)      * 64 * 128, h0 + nl * HID, 64, off, lane);
      ar  = mma_pass(ar,  Phh1 + (size_t)(t)      * 64 * 128, h1 + nl * HID, 64, off, lane);
      az  = mma_pass(az,  Pih1 + (size_t)(32 + t) * 64 * 128, h0 + nl * HID, 64, off, lane);
      az  = mma_pass(az,  Phh1 + (size_t)(32 + t) * 64 * 128, h1 + nl * HID, 64, off, lane);
      ain = mma_pass(ain, Pih1 + (size_t)(64 + t) * 64 * 128, h0 + nl * HID, 64, off, lane);
      ahn = mma_pass(ahn, Phh1 + (size_t)(64 + t) * 64 * 128, h1 + nl * HID, 64, off, lane);
      float bri = bih1[t * 16 + nl] + bhh1[t * 16 + nl];
      float bzi = bih1[HID + t * 16 + nl] + bhh1[HID + t * 16 + nl];
      float bni = bih1[2 * HID + t * 16 + nl];
      float bnh = bhh1[2 * HID + t * 16 + nl];
#pragma unroll
      for (int e = 0; e < 8; ++e) {
        float r  = sigf(ar[e] + bri);
        float z  = sigf(az[e] + bzi);
        float nn = tanhf(ain[e] + bni + r * (ahn[e] + bnh));
        float ho = h1[(e + hi8) * HID + t * 16 + nl];
        hnew[j][e] = (1.f - z) * nn + z * ho;
      }
    }
    __syncthreads();
#pragma unroll
    for (int j = 0; j < 4; ++j) {
      int t = wave + j * 8;
#pragma unroll
      for (int e = 0; e < 8; ++e)
        h1[(e + hi8) * HID + t * 16 + nl] = hnew[j][e];
    }
    __syncthreads();

    // ================= FC logits =================
#pragma unroll
    for (int j = 0; j < 2; ++j) {
      int t = wave + j * 8;  // [0,16)
      v8f acc = {};
      acc = mma_pass(acc, Pfc + (size_t)t * 64 * 128, h1 + nl * HID, 64, off, lane);
      float bias = bfc[t * 16 + nl];
#pragma unroll
      for (int e = 0; e < 8; ++e)
        logits[(e + hi8) * VOCAB + t * 16 + nl] = acc[e] + bias;
    }
    __syncthreads();

    // ================= argmax (first-max, matches jnp.argmax) =================
    {
      int row = tid >> 4, sub = tid & 15;
      float best = -1e30f; int bi = 0;
#pragma unroll
      for (int k = 0; k < 16; ++k) {
        float f = logits[row * VOCAB + sub * 16 + k];
        if (f > best) { best = f; bi = sub * 16 + k; }
      }
#pragma unroll
      for (int d = 8; d >= 1; d >>= 1) {
        float ob = __shfl_xor(best, d, 16);
        int  obi = __shfl_xor(bi, d, 16);
        if (ob > best || (ob == best && obi < bi)) { best = ob; bi = obi; }
      }
      if (sub == 0) idxm[row] = bi;
    }
    __syncthreads();

    // ================= one-hot output write =================
#pragma unroll 1
    for (int m = 0; m < 16; ++m) {
      float v = (tid == idxm[m]) ? 1.0f : 0.0f;
      out[((size_t)(rowbase + m) * T_STEPS + step) * VOCAB + tid] = v;
    }
    __syncthreads();
  }
}

// ---------------------------------------------------------------------------
extern "C" void kernel_launch(void* const* d_in, const int* in_sizes, int n_in,
                              void* d_out, int out_size, void* d_ws, size_t ws_size,
                              hipStream_t stream) {
  (void)in_sizes; (void)n_in; (void)out_size; (void)ws_size;
  const float* latent = (const float*)d_in[0];
  // d_in[1] = enthalpy (unused on freerun path)
  const float* w_ih0 = (const float*)d_in[2];
  const float* w_hh0 = (const float*)d_in[3];
  const float* b_ih0 = (const float*)d_in[4];
  const float* b_hh0 = (const float*)d_in[5];
  const float* w_ih1 = (const float*)d_in[6];   // [1,3H,H]
  const float* w_hh1 = (const float*)d_in[7];
  const float* b_ih1 = (const float*)d_in[8];
  const float* b_hh1 = (const float*)d_in[9];
  const float* w_fc  = (const float*)d_in[10];
  const float* b_fc  = (const float*)d_in[11];
  float* ws = (float*)d_ws;

  size_t o = 0;
  float* gi0ws = ws + o; o += (size_t)64 * 16 * G3;     // 1.57M floats
  float* Pih0  = ws + o; o += (size_t)96 * 32 * 128;    // 0.39M
  float* WohT  = ws + o; o += (size_t)VOCAB * G3;       // 0.39M
  float* Phh0  = ws + o; o += (size_t)96 * 64 * 128;    // 0.79M
  float* Pih1  = ws + o; o += (size_t)96 * 64 * 128;
  float* Phh1  = ws + o; o += (size_t)96 * 64 * 128;
  float* Pfc   = ws + o; o += (size_t)16 * 64 * 128;    // total ~19.4 MB

  auto lp = [&](const float* W, float* dst, int nt, int ncp) {
    int total = nt * ncp * 128;
    pack_frag<<<(total + 255) / 256, 256, 0, stream>>>(W, dst, nt, ncp, 512, 0);
  };
  lp(w_ih0, Pih0, 96, 32);   // latent columns only (K=256)
  lp(w_hh0, Phh0, 96, 64);
  lp(w_ih1, Pih1, 96, 64);
  lp(w_hh1, Phh1, 96, 64);
  lp(w_fc,  Pfc,  16, 64);
  transpose_oh<<<(VOCAB * G3 + 255) / 256, 256, 0, stream>>>(w_ih0, WohT);

  size_t smem = (size_t)(2 * 16 * HID + 16 * VOCAB) * sizeof(float) + 16 * sizeof(int);
  (void)hipFuncSetAttribute(reinterpret_cast<const void*>(decoder_main),
                            hipFuncAttributeMaxDynamicSharedMemorySize, (int)smem);
  decoder_main<<<64, 256, smem, stream>>>(latent, b_ih0, b_hh0, b_ih1, b_hh1, b_fc,
                                          Pih0, WohT, Phh0, Pih1, Phh1, Pfc,
                                          gi0ws, (float*)d_out);
}